// PaiNNMessageBlock_11347303596608
// MI455X (gfx1250) — compile-verified
//
#include <hip/hip_runtime.h>
#include <hip/hip_bf16.h>
#include <math.h>

#define N_NODES 10000
#define N_EDGES 320000
#define FDIM    128
#define N_RBF   20

typedef __attribute__((ext_vector_type(2))) float v2f;
typedef __attribute__((ext_vector_type(8))) float v8f;

// ---------------------------------------------------------------------------
// WMMA fragment layouts (CDNA5, wave32, V_WMMA_F32_16X16X4_F32):
//   A (16x4, MxK):  lane L -> m = L%16, h = L/16;  a.x = A[m][k+2h], a.y = A[m][k+2h+1]
//   B (4x16, KxN):  lane L -> n = L%16, h = L/16;  b.x = B[k+2h][n], b.y = B[k+2h+1][n]
//   C/D (16x16):    lane L -> n = L%16, h = L/16;  c[i] = C[i+8h][n]
// ---------------------------------------------------------------------------

__device__ __forceinline__ v8f wmma_f32(v2f a, v2f b, v8f c) {
  return __builtin_amdgcn_wmma_f32_16x16x4_f32(false, a, false, b, (short)0, c,
                                               false, false);
}

// ---------------------------------------------------------------------------
// Kernel 1: phi = silu(S @ W1 + b1) @ W2 + b2        for a 16-node row tile
// 8 waves / block. Wave w: H column tile w (stage 1), phi column tiles
// 3w..3w+2 (stage 2). H staged in LDS with padded stride 132.
// ---------------------------------------------------------------------------
__global__ void __launch_bounds__(256)
node_phi_kernel(const float* __restrict__ S, const float* __restrict__ W1,
                const float* __restrict__ b1, const float* __restrict__ W2,
                const float* __restrict__ b2, float* __restrict__ phi) {
  __shared__ float Hs[16 * 132];

  const int wave = threadIdx.x >> 5;
  const int lane = threadIdx.x & 31;
  const int h    = lane >> 4;      // half-wave
  const int mn   = lane & 15;      // row (A) / col (B, C/D) within tile
  const int n0   = blockIdx.x * 16;

  // ---- stage 1: H = silu(S @ W1 + b1), this wave's 16-col tile ----
  {
    const int col = wave * 16 + mn;
    v8f c = {};
    #pragma unroll
    for (int k = 0; k < FDIM; k += 4) {
      v2f a, b;
      const int ka = k + 2 * h;
      a.x = S[(n0 + mn) * FDIM + ka];
      a.y = S[(n0 + mn) * FDIM + ka + 1];
      b.x = W1[ka * FDIM + col];
      b.y = W1[(ka + 1) * FDIM + col];
      c = wmma_f32(a, b, c);
    }
    const float bb = b1[col];
    #pragma unroll
    for (int i = 0; i < 8; ++i) {
      const float x = c[i] + bb;
      Hs[(i + 8 * h) * 132 + col] = x / (1.0f + __expf(-x));  // silu
    }
  }
  __syncthreads();

  // ---- stage 2: phi = H @ W2 + b2, three 16-col tiles per wave ----
  #pragma unroll
  for (int s = 0; s < 3; ++s) {
    const int col = (wave * 3 + s) * 16 + mn;
    v8f c = {};
    #pragma unroll
    for (int k = 0; k < FDIM; k += 4) {
      v2f a, b;
      const int ka = k + 2 * h;
      a.x = Hs[mn * 132 + ka];
      a.y = Hs[mn * 132 + ka + 1];
      b.x = W2[ka * 384 + col];
      b.y = W2[(ka + 1) * 384 + col];
      c = wmma_f32(a, b, c);
    }
    const float bb = b2[col];
    #pragma unroll
    for (int i = 0; i < 8; ++i)
      phi[(size_t)(n0 + i + 8 * h) * 384 + col] = c[i] + bb;
  }
}

// ---------------------------------------------------------------------------
// Kernel 2: per 16-edge tile:
//   Wg   = (rbf @ Wr + br) * rel_dist_cut        (WMMA, K=20)
//   pw   = phi[idx_j] * Wg                       (fused gather, into LDS)
//   scatter: sres += pw_ss ; vres += V[idx_j]*pw_vv + pw_vs*rel_dir ; counts++
// ---------------------------------------------------------------------------
__global__ void __launch_bounds__(256)
edge_kernel(const int* __restrict__ idx_i, const int* __restrict__ idx_j,
            const float* __restrict__ rel_dir,
            const float* __restrict__ rel_dist_cut,
            const float* __restrict__ rbf, const float* __restrict__ Wr,
            const float* __restrict__ br, const float* __restrict__ phi,
            const float* __restrict__ V, float* __restrict__ sres,
            float* __restrict__ vres, float* __restrict__ counts) {
  __shared__ float pw[16 * 388];   // 16 edges x 384 cols, padded stride

  const int wave = threadIdx.x >> 5;
  const int lane = threadIdx.x & 31;
  const int h    = lane >> 4;
  const int mn   = lane & 15;
  const int e0   = blockIdx.x * 16;

  // ---- stage A: WMMA Wg + fused gather-multiply, 3 col tiles per wave ----
  #pragma unroll
  for (int s = 0; s < 3; ++s) {
    const int col = (wave * 3 + s) * 16 + mn;
    v8f c = {};
    #pragma unroll
    for (int k = 0; k < N_RBF; k += 4) {
      v2f a, b;
      const int ka = k + 2 * h;
      a.x = rbf[(e0 + mn) * N_RBF + ka];
      a.y = rbf[(e0 + mn) * N_RBF + ka + 1];
      b.x = Wr[ka * 384 + col];
      b.y = Wr[(ka + 1) * 384 + col];
      c = wmma_f32(a, b, c);
    }
    const float bb = br[col];
    #pragma unroll
    for (int i = 0; i < 8; ++i) {
      const int m = i + 8 * h;
      const int e = e0 + m;
      const float wg = (c[i] + bb) * rel_dist_cut[e];
      const float pj = phi[(size_t)idx_j[e] * 384 + col];
      pw[m * 388 + col] = pj * wg;
    }
  }
  __syncthreads();

  // ---- stage B: scatter-add into node accumulators (L2-resident) ----
  for (int it = threadIdx.x; it < 16 * FDIM; it += 256) {
    const int el = it >> 7;        // edge within tile
    const int f  = it & (FDIM - 1);
    const int e  = e0 + el;
    const int ni = idx_i[e];
    const int nj = idx_j[e];
    const float vv = pw[el * 388 + f];
    const float ss = pw[el * 388 + FDIM + f];
    const float vs = pw[el * 388 + 2 * FDIM + f];
    atomicAdd(&sres[(size_t)ni * FDIM + f], ss);
    const float d0 = rel_dir[e * 3 + 0];
    const float d1 = rel_dir[e * 3 + 1];
    const float d2 = rel_dir[e * 3 + 2];
    const float* vf = &V[((size_t)nj * FDIM + f) * 3];
    float* vo = &vres[((size_t)ni * FDIM + f) * 3];
    atomicAdd(&vo[0], vf[0] * vv + vs * d0);
    atomicAdd(&vo[1], vf[1] * vv + vs * d1);
    atomicAdd(&vo[2], vf[2] * vv + vs * d2);
  }
  if (threadIdx.x < 16)
    atomicAdd(&counts[idx_i[e0 + threadIdx.x]], 1.0f);
}

// ---------------------------------------------------------------------------
// Kernel 3: out = inputs + residual / counts
// ---------------------------------------------------------------------------
__global__ void __launch_bounds__(256)
finalize_kernel(const float* __restrict__ S, const float* __restrict__ V,
                const float* __restrict__ sres, const float* __restrict__ vres,
                const float* __restrict__ counts, float* __restrict__ out) {
  const int NS = N_NODES * FDIM;
  const int NV = N_NODES * FDIM * 3;
  const int t = blockIdx.x * blockDim.x + threadIdx.x;
  if (t < NS) {
    const int n = t >> 7;  // /128
    out[t] = S[t] + sres[t] / counts[n];
  } else if (t < NS + NV) {
    const int v = t - NS;
    const int n = v / 384;
    out[t] = V[v] + vres[v] / counts[n];
  }
}

// ---------------------------------------------------------------------------
extern "C" void kernel_launch(void* const* d_in, const int* in_sizes, int n_in,
                              void* d_out, int out_size, void* d_ws,
                              size_t ws_size, hipStream_t stream) {
  const int*   idx_i        = (const int*)d_in[0];
  const int*   idx_j        = (const int*)d_in[1];
  const float* rel_dir      = (const float*)d_in[2];
  const float* rel_dist_cut = (const float*)d_in[3];
  const float* rbf          = (const float*)d_in[4];
  const float* S            = (const float*)d_in[5];
  const float* V            = (const float*)d_in[6];
  const float* W1           = (const float*)d_in[7];
  const float* b1           = (const float*)d_in[8];
  const float* W2           = (const float*)d_in[9];
  const float* b2           = (const float*)d_in[10];
  const float* Wr           = (const float*)d_in[11];
  const float* br           = (const float*)d_in[12];
  float*       out          = (float*)d_out;

  // workspace layout (floats): phi | sres | vres | counts  (~35.9 MB)
  float* phi    = (float*)d_ws;
  float* sres   = phi + (size_t)N_NODES * 384;
  float* vres   = sres + (size_t)N_NODES * FDIM;
  float* cnts   = vres + (size_t)N_NODES * FDIM * 3;

  const size_t acc_bytes =
      ((size_t)N_NODES * FDIM + (size_t)N_NODES * FDIM * 3 + N_NODES) *
      sizeof(float);
  hipMemsetAsync(sres, 0, acc_bytes, stream);

  node_phi_kernel<<<N_NODES / 16, 256, 0, stream>>>(S, W1, b1, W2, b2, phi);
  edge_kernel<<<N_EDGES / 16, 256, 0, stream>>>(idx_i, idx_j, rel_dir,
                                                rel_dist_cut, rbf, Wr, br, phi,
                                                V, sres, vres, cnts);
  const int total = N_NODES * FDIM * 4;
  finalize_kernel<<<(total + 255) / 256, 256, 0, stream>>>(S, V, sres, vres,
                                                           cnts, out);
}